// HungarianMatcher_6390911336916
// MI455X (gfx1250) — compile-verified
//
#include <hip/hip_runtime.h>

// HungarianMatcher cost matrix, store-bandwidth-bound (128 MiB fp32 output).
// C[r,t] = 10*(|qx-tx|+|qy-ty|) + 2*(pos(r,id_t) - neg(r,id_t))
// Class-gather term computed as exact f32 WMMA: D = Delta[16x2(pad4)] x OneHot[2(pad4)x16] + C(10*l1)

typedef __attribute__((ext_vector_type(2))) float v2f;
typedef __attribute__((ext_vector_type(8))) float v8f;

constexpr int TC = 4096;            // bs*T (columns)
constexpr int RQ = 8192;            // bs*Q (rows)
constexpr int ROWS_PER_BLK = 32;    // two stacked 16x16 WMMA tiles per wave
constexpr int COLS_PER_BLK = 128;   // 8 waves x 16 cols

__global__ __launch_bounds__(256) void matcher_cost_wmma(
    const float* __restrict__ pred_class,   // [RQ, 2] logits
    const float* __restrict__ pred_points,  // [RQ, 2]
    const int*   __restrict__ gt_class,     // [TC]
    const float* __restrict__ gt_points,    // [TC, 2]
    float* __restrict__ out)                // [RQ, TC]
{
    __shared__ float4 srow[ROWS_PER_BLK];   // (qx, qy, 2*delta0, 2*delta1)
    __shared__ float4 scol[COLS_PER_BLK];   // (tx, ty, [id==0], [id==1])

    const int tid  = threadIdx.x;
    const int row0 = blockIdx.y * ROWS_PER_BLK;
    const int col0 = blockIdx.x * COLS_PER_BLK;

    // ---- stage row operands (32 rows/block): focal class deltas + points ----
    if (tid < ROWS_PER_BLK) {
        const int r = row0 + tid;
        const float qx = pred_points[2 * r];
        const float qy = pred_points[2 * r + 1];
        float d0 = 0.0f, d1 = 0.0f;
        #pragma unroll
        for (int cc = 0; cc < 2; ++cc) {
            const float x   = pred_class[2 * r + cc];
            const float p   = 1.0f / (1.0f + __expf(-x));
            const float omp = 1.0f - p;
            const float pos = 0.25f * omp * omp * (-__logf(p   + 1e-8f));
            const float neg = 0.75f * p   * p   * (-__logf(omp + 1e-8f));
            const float d   = 2.0f * (pos - neg);   // CLASS_WEIGHT folded in
            if (cc == 0) d0 = d; else d1 = d;
        }
        srow[tid] = make_float4(qx, qy, d0, d1);
    }
    // ---- stage column operands (128 cols/block): target points + one-hot id ----
    if (tid >= 128) {
        const int j = tid - 128;
        const int c = col0 + j;
        const int id = gt_class[c];
        scol[j] = make_float4(gt_points[2 * c], gt_points[2 * c + 1],
                              (id == 0) ? 1.0f : 0.0f,
                              (id == 1) ? 1.0f : 0.0f);
    }
    __syncthreads();

    // ---- each wave: one 16-col strip, two 16x16 WMMA tiles (rows 0-15, 16-31) ----
    const int wave = tid >> 5;
    const int lane = tid & 31;
    const int n    = lane & 15;    // M (for A) / N (for B,C,D)
    const int half = lane >> 4;    // 0: K=0..1 / rows 0..7 ; 1: K=2..3 / rows 8..15
    const int jn   = wave * 16;

    // B: 4x16 f32, B[0,n]=[id_n==0], B[1,n]=[id_n==1], K=2,3 zero (shared by both tiles)
    const float4 colD = scol[jn + n];
    v2f b;
    b.x = half ? 0.0f : colD.z;
    b.y = half ? 0.0f : colD.w;

    const uint32_t cidx = (uint32_t)(col0 + jn + n);

    #pragma unroll
    for (int t = 0; t < 2; ++t) {
        const int rbase = t * 16;

        // A: 16x4 f32, A[m,0]=2*delta0[m], A[m,1]=2*delta1[m], K=2,3 zero
        const float4 rowA = srow[rbase + n];
        v2f a;
        a.x = half ? 0.0f : rowA.z;
        a.y = half ? 0.0f : rowA.w;

        // C: 10 * L1 distance; VGPR v -> M = v + 8*half, N = n
        v8f c;
        #pragma unroll
        for (int v = 0; v < 8; ++v) {
            const float4 rw = srow[rbase + v + 8 * half];
            c[v] = 10.0f * (fabsf(rw.x - colD.x) + fabsf(rw.y - colD.y));
        }

        // D = A x B + C   (exact fp32)
        v8f d = __builtin_amdgcn_wmma_f32_16x16x4_f32(
            /*neg_a=*/false, a, /*neg_b=*/false, b,
            /*c_mod=*/(short)0, c, /*reuse_a=*/false, /*reuse_b=*/false);

        // 32-bit indexing, compile-time row stride -> saddr + imm-offset stores
        const uint32_t idx = (uint32_t)(row0 + rbase + 8 * half) * (uint32_t)TC + cidx;
        #pragma unroll
        for (int v = 0; v < 8; ++v)
            out[idx + (uint32_t)(v * TC)] = d[v];
    }
}

extern "C" void kernel_launch(void* const* d_in, const int* in_sizes, int n_in,
                              void* d_out, int out_size, void* d_ws, size_t ws_size,
                              hipStream_t stream) {
    const float* pred_class  = (const float*)d_in[0];  // [8,1024,2] f32
    const float* pred_points = (const float*)d_in[1];  // [8,1024,2] f32
    const int*   gt_class    = (const int*)d_in[2];    // [8,512] int32
    const float* gt_points   = (const float*)d_in[3];  // [8,512,2] f32
    float* out = (float*)d_out;                        // [8,1024,4096] f32

    dim3 grid(TC / COLS_PER_BLK, RQ / ROWS_PER_BLK);   // 32 x 256 blocks
    matcher_cost_wmma<<<grid, 256, 0, stream>>>(
        pred_class, pred_points, gt_class, gt_points, out);
}